// VisitNeuralODE_62122406969959
// MI455X (gfx1250) — compile-verified
//
#include <hip/hip_runtime.h>
#include <hip/hip_bf16.h>
#include <math.h>

// Problem sizes (fixed by the reference): T=128, D=1024, H=2048.
#define T_N   128
#define D_N   1024
#define H_N   2048
#define G3H   (3 * H_N)

// Persistent-kernel geometry: 256 blocks x 256 threads = 2048 wave32 waves.
#define NBLK   256
#define NTHR   256
#define NWAVES ((NBLK * NTHR) / 32)   // 2048

typedef __attribute__((ext_vector_type(2))) float v2f;
typedef __attribute__((ext_vector_type(8))) float v8f;

// Dormand-Prince 5(4) tableau
__constant__ float c_dA[5][5] = {
    {0.2f, 0.f, 0.f, 0.f, 0.f},
    {0.075f, 0.225f, 0.f, 0.f, 0.f},
    {(float)(44.0/45.0), (float)(-56.0/15.0), (float)(32.0/9.0), 0.f, 0.f},
    {(float)(19372.0/6561.0), (float)(-25360.0/2187.0), (float)(64448.0/6561.0),
     (float)(-212.0/729.0), 0.f},
    {(float)(9017.0/3168.0), (float)(-355.0/33.0), (float)(46732.0/5247.0),
     (float)(49.0/176.0), (float)(-5103.0/18656.0)}};
__constant__ float c_dB[6] = {
    (float)(35.0/384.0), 0.f, (float)(500.0/1113.0), (float)(125.0/192.0),
    (float)(-2187.0/6784.0), (float)(11.0/84.0)};

__device__ __forceinline__ float warp_reduce(float v) {
  #pragma unroll
  for (int off = 16; off > 0; off >>= 1) v += __shfl_xor(v, off, 32);
  return v;
}

__device__ __forceinline__ float sigmoidf_(float x) {
  return 1.f / (1.f + __expf(-x));
}

// Monotonic-counter grid barrier. Counter region is memset to 0 each launch.
__device__ __forceinline__ void gbar(int* cnt, int& epoch) {
  __syncthreads();
  if (threadIdx.x == 0) {
    __hip_atomic_fetch_add(cnt, 1, __ATOMIC_ACQ_REL, __HIP_MEMORY_SCOPE_AGENT);
    epoch += 1;
    const int target = epoch * NBLK;
    while (__hip_atomic_load(cnt, __ATOMIC_ACQUIRE, __HIP_MEMORY_SCOPE_AGENT) < target) {
      __builtin_amdgcn_s_sleep(2);
    }
  }
  __syncthreads();
}

// y[r] = (tanh?)(dot(W[r,:], u) + bias[r]) for r partitioned over all waves.
// u points into LDS; W rows read as coalesced lane-strided float4.
__device__ __forceinline__ void gemv_phase(const float* __restrict__ W,
                                           const float* __restrict__ bias,
                                           const float* __restrict__ u,
                                           float* __restrict__ y,
                                           int rows, int K, bool act_tanh) {
  const int gtid = blockIdx.x * NTHR + threadIdx.x;
  const int wave = gtid >> 5;
  const int lane = gtid & 31;
  const int rpw  = rows / NWAVES;     // 1 for H rows
  const int K4   = K >> 2;
  const float4* U4 = (const float4*)u;
  for (int rr = 0; rr < rpw; ++rr) {
    const int r = wave * rpw + rr;
    const float4* Wr = (const float4*)(W + (size_t)r * (size_t)K);
    float acc = 0.f;
    #pragma unroll 4
    for (int kk = lane; kk < K4; kk += 32) {
      float4 w = Wr[kk];
      float4 uu = U4[kk];
      acc += w.x * uu.x + w.y * uu.y + w.z * uu.z + w.w * uu.w;
    }
    acc = warp_reduce(acc);
    if (lane == 0) {
      float v = acc + bias[r];
      y[r] = act_tanh ? tanhf(v) : v;
    }
  }
}

// Fused GRU GEMV phase: waves 0..1023 compute gi = Wih x + bih (K=1024, u=x in
// LDS); waves 1024..2047 compute gh = Whh h + bhh (K=2048, u=h replicated LDS).
__device__ __forceinline__ void gru_gemv_phase(
    const float* __restrict__ wih, const float* __restrict__ bih,
    const float* __restrict__ whh, const float* __restrict__ bhh,
    const float* __restrict__ xlds, const float* __restrict__ hlds,
    float* __restrict__ gi, float* __restrict__ gh) {
  const int gtid = blockIdx.x * NTHR + threadIdx.x;
  const int wave = gtid >> 5;
  const int lane = gtid & 31;
  const bool is_gh = (wave >= NWAVES / 2);
  const int  wv    = is_gh ? (wave - NWAVES / 2) : wave;
  const int  K     = is_gh ? H_N : D_N;
  const int  K4    = K >> 2;
  const float4* U4 = (const float4*)(is_gh ? hlds : xlds);
  const float* W   = is_gh ? whh : wih;
  const float* B   = is_gh ? bhh : bih;
  float* Y         = is_gh ? gh : gi;
  #pragma unroll
  for (int rr = 0; rr < 6; ++rr) {           // 6144 rows / 1024 waves
    const int r = wv * 6 + rr;
    const float4* Wr = (const float4*)(W + (size_t)r * (size_t)K);
    float acc = 0.f;
    #pragma unroll 4
    for (int kk = lane; kk < K4; kk += 32) {
      float4 w = Wr[kk];
      float4 uu = U4[kk];
      acc += w.x * uu.x + w.y * uu.y + w.z * uu.z + w.w * uu.w;
    }
    acc = warp_reduce(acc);
    if (lane == 0) Y[r] = acc + B[r];
  }
}

// ---------------------------------------------------------------------------
// Persistent kernel: whole T-loop recurrence (ODE dopri5 + GRU), emits Hmat.
// Hidden state h lives replicated in per-block LDS; all blocks recompute the
// (identical, fixed FP order) h updates from globally-synced k/gi/gh vectors.
// ---------------------------------------------------------------------------
__global__ __launch_bounds__(NTHR) void seq_recurrence_kernel(
    const float* __restrict__ visit_x, const float* __restrict__ delta_t,
    const float* __restrict__ pin_w,  const float* __restrict__ pin_b,
    const float* __restrict__ ow1,    const float* __restrict__ ob1,
    const float* __restrict__ ow2,    const float* __restrict__ ob2,
    const float* __restrict__ wih,    const float* __restrict__ whh,
    const float* __restrict__ bih,    const float* __restrict__ bhh,
    float* __restrict__ tv, float* __restrict__ kbuf,
    float* __restrict__ gi, float* __restrict__ gh, float* __restrict__ Hmat,
    int* cnt) {
  __shared__ __align__(16) float uld[H_N];   // stage input / x staging
  __shared__ __align__(16) float hld[H_N];   // replicated hidden state
  int epoch = 0;
  const int tid = threadIdx.x;

  // h0 = proj_in_w @ x0 + b  (staged through tv, then replicated into LDS)
  for (int e = tid; e < D_N; e += NTHR) uld[e] = visit_x[e];
  __syncthreads();
  gemv_phase(pin_w, pin_b, uld, tv, H_N, D_N, false);
  gbar(cnt, epoch);
  for (int e = tid; e < H_N; e += NTHR) {
    float v = tv[e];
    hld[e] = v;
    if (blockIdx.x == 0) Hmat[e] = v;
  }

  for (int t = 1; t < T_N; ++t) {
    const float dt = delta_t[t] * 0.125f;  // dt_total / N_ODE_STEPS

    for (int step = 0; step < 8; ++step) {
      for (int j = 0; j < 6; ++j) {
        // u = h + dt * sum_i A[j-1][i] * k_i  (redundant per block, h in LDS)
        __syncthreads();
        for (int e = tid; e < H_N; e += NTHR) {
          float s = hld[e];
          if (j > 0) {
            float a = 0.f;
            for (int i2 = 0; i2 < j; ++i2) a += c_dA[j - 1][i2] * kbuf[i2 * H_N + e];
            s += dt * a;
          }
          uld[e] = s;
        }
        __syncthreads();
        gemv_phase(ow1, ob1, uld, tv, H_N, H_N, true);   // tv = tanh(W1 u + b1)
        gbar(cnt, epoch);
        for (int e = tid; e < H_N; e += NTHR) uld[e] = tv[e];
        __syncthreads();
        gemv_phase(ow2, ob2, uld, kbuf + j * H_N, H_N, H_N, false);  // k_j
        gbar(cnt, epoch);
      }
      // h += dt * sum_i B[i] k_i  (redundant per block into LDS; no barrier)
      for (int e = tid; e < H_N; e += NTHR) {
        float s = 0.f;
        #pragma unroll
        for (int i2 = 0; i2 < 6; ++i2) s += c_dB[i2] * kbuf[i2 * H_N + e];
        hld[e] += dt * s;
      }
      // next stage build starts with __syncthreads(); block-local order OK
    }

    // GRU: fused gi/gh GEMV phase (x staged in uld, h already in hld)
    __syncthreads();
    for (int e = tid; e < D_N; e += NTHR) uld[e] = visit_x[(size_t)t * D_N + e];
    __syncthreads();
    gru_gemv_phase(wih, bih, whh, bhh, uld, hld, gi, gh);
    gbar(cnt, epoch);
    // Combine redundantly per block; block 0 also writes Hmat[t].
    for (int e = tid; e < H_N; e += NTHR) {
      float r = sigmoidf_(gi[e] + gh[e]);
      float z = sigmoidf_(gi[H_N + e] + gh[H_N + e]);
      float n = tanhf(gi[2 * H_N + e] + r * gh[2 * H_N + e]);
      float hn = (1.f - z) * n + z * hld[e];
      hld[e] = hn;
      if (blockIdx.x == 0) Hmat[(size_t)t * H_N + e] = hn;
    }
    // next visit's stage build begins with __syncthreads()
  }
}

// ---------------------------------------------------------------------------
// WMMA f32 GEMM: out[M,N] = act(A[M,K] @ Wt[N,K]^T + bias), one 16x16 tile
// per wave via V_WMMA_F32_16X16X4_F32 (lossless f32 matrix path).
// A layout: lanes 0..15 -> M rows with K,K+1; lanes 16..31 carry K+2,K+3.
// B symmetric with N rows of Wt. EXEC is all-ones (exact tile grid).
// ---------------------------------------------------------------------------
template <int ACT>
__global__ __launch_bounds__(128) void gemm16_wmma(
    const float* __restrict__ A, const float* __restrict__ Wt,
    const float* __restrict__ bias, float* __restrict__ out,
    int M, int N, int K) {
  const int gw   = (blockIdx.x * blockDim.x + threadIdx.x) >> 5;
  const int lane = threadIdx.x & 31;
  const int tN   = N >> 4;
  const int tm   = (gw / tN) << 4;
  const int tn   = (gw % tN) << 4;
  const int mn   = lane & 15;          // M index for A, N index for B/C
  const int kh   = (lane >> 4) << 1;   // K sub-offset: 0 (lanes 0-15), 2 (16-31)

  const float* Arow = A  + (size_t)(tm + mn) * (size_t)K + kh;
  const float* Brow = Wt + (size_t)(tn + mn) * (size_t)K + kh;

  v8f acc = {0.f, 0.f, 0.f, 0.f, 0.f, 0.f, 0.f, 0.f};
  #pragma unroll 8
  for (int k0 = 0; k0 < K; k0 += 4) {
    v2f a, b;
    a.x = Arow[k0];     a.y = Arow[k0 + 1];
    b.x = Brow[k0];     b.y = Brow[k0 + 1];
    acc = __builtin_amdgcn_wmma_f32_16x16x4_f32(
        /*neg_a=*/false, a, /*neg_b=*/false, b,
        /*c_mod=*/(short)0, acc, /*reuse_a=*/false, /*reuse_b=*/false);
  }

  const int mo = (lane >> 4) << 3;     // row offset 0 or 8 in C layout
  const float bv = bias[tn + mn];
  #pragma unroll
  for (int r = 0; r < 8; ++r) {
    float v = acc[r] + bv;
    if (ACT == 1) v = tanhf(v);
    out[(size_t)(tm + r + mo) * (size_t)N + (tn + mn)] = v;
  }
}

// scores[m] = S1[m,:] . att_w2 + b2   (one wave per row, deterministic order)
__global__ __launch_bounds__(128) void scores_kernel(
    const float* __restrict__ S1, const float* __restrict__ w2,
    const float* __restrict__ b2, float* __restrict__ scores) {
  const int wave = (blockIdx.x * blockDim.x + threadIdx.x) >> 5;
  const int lane = threadIdx.x & 31;
  const float* row = S1 + (size_t)wave * H_N;
  float acc = 0.f;
  for (int k = lane; k < H_N; k += 32) acc += row[k] * w2[k];
  acc = warp_reduce(acc);
  if (lane == 0) scores[wave] = acc + b2[0];
}

// softmax over T=128 scores -> alpha (single block, deterministic)
__global__ __launch_bounds__(T_N) void softmax_kernel(
    const float* __restrict__ scores, float* __restrict__ alpha) {
  __shared__ float sm[T_N];
  __shared__ float mxs;
  const int i = threadIdx.x;
  const float s = scores[i];
  sm[i] = s;
  __syncthreads();
  for (int off = T_N / 2; off > 0; off >>= 1) {
    if (i < off) sm[i] = fmaxf(sm[i], sm[i + off]);
    __syncthreads();
  }
  if (i == 0) mxs = sm[0];
  __syncthreads();
  const float e = __expf(s - mxs);
  sm[i] = e;
  __syncthreads();
  for (int off = T_N / 2; off > 0; off >>= 1) {
    if (i < off) sm[i] += sm[i + off];
    __syncthreads();
  }
  alpha[i] = e / sm[0];
}

extern "C" void kernel_launch(void* const* d_in, const int* in_sizes, int n_in,
                              void* d_out, int out_size, void* d_ws, size_t ws_size,
                              hipStream_t stream) {
  (void)in_sizes; (void)n_in; (void)out_size; (void)ws_size;

  const float* visit_x   = (const float*)d_in[0];
  const float* delta_t   = (const float*)d_in[1];
  const float* proj_in_w = (const float*)d_in[2];
  const float* proj_in_b = (const float*)d_in[3];
  const float* ode_w1    = (const float*)d_in[4];
  const float* ode_b1    = (const float*)d_in[5];
  const float* ode_w2    = (const float*)d_in[6];
  const float* ode_b2    = (const float*)d_in[7];
  const float* gru_wih   = (const float*)d_in[8];
  const float* gru_whh   = (const float*)d_in[9];
  const float* gru_bih   = (const float*)d_in[10];
  const float* gru_bhh   = (const float*)d_in[11];
  const float* proj_out_w= (const float*)d_in[12];
  const float* proj_out_b= (const float*)d_in[13];
  const float* att_w1    = (const float*)d_in[14];
  const float* att_b1    = (const float*)d_in[15];
  const float* att_w2    = (const float*)d_in[16];
  const float* att_b2    = (const float*)d_in[17];

  // Workspace layout (~2.2 MB)
  char*  wsb   = (char*)d_ws;
  int*   cnt   = (int*)wsb;                 // 256B barrier counter region
  float* f     = (float*)(wsb + 256);
  float* tv    = f;                          // [H]   (also h0 staging)
  float* kbuf  = tv + H_N;                   // [6*H]
  float* gi    = kbuf + 6 * H_N;             // [3H]
  float* gh    = gi + G3H;                   // [3H]
  float* Hmat  = gh + G3H;                   // [T*H]
  float* S1    = Hmat + (size_t)T_N * H_N;   // [T*H]
  float* scr   = S1 + (size_t)T_N * H_N;     // [T]

  float* out_x     = (float*)d_out;              // [T*D]
  float* out_alpha = out_x + (size_t)T_N * D_N;  // [T]

  hipMemsetAsync(cnt, 0, 256, stream);

  seq_recurrence_kernel<<<NBLK, NTHR, 0, stream>>>(
      visit_x, delta_t, proj_in_w, proj_in_b, ode_w1, ode_b1, ode_w2, ode_b2,
      gru_wih, gru_whh, gru_bih, gru_bhh, tv, kbuf, gi, gh, Hmat, cnt);

  // S1 = tanh(Hmat @ att_w1^T + att_b1)  : M=128, N=2048, K=2048 -> 1024 tiles
  gemm16_wmma<1><<<256, 128, 0, stream>>>(Hmat, att_w1, att_b1, S1,
                                          T_N, H_N, H_N);
  // scores = S1 @ att_w2 + b2
  scores_kernel<<<32, 128, 0, stream>>>(S1, att_w2, att_b2, scr);
  // visit_x_new = Hmat @ proj_out_w^T + proj_out_b : M=128, N=1024, K=2048
  gemm16_wmma<0><<<128, 128, 0, stream>>>(Hmat, proj_out_w, proj_out_b, out_x,
                                          T_N, D_N, H_N);
  // alpha = softmax(scores)
  softmax_kernel<<<1, T_N, 0, stream>>>(scr, out_alpha);
}